// AntisymmetricExpGenerator_12214886989921
// MI455X (gfx1250) — compile-verified
//
#include <hip/hip_runtime.h>

typedef __attribute__((ext_vector_type(2))) float v2f;
typedef __attribute__((ext_vector_type(8))) float v8f;

#define HD   2048          // H_DIM
#define BSZ  512           // batch
#define UD   1024          // U_DIM
#define DUD  512           // DU_DIM
#define KIN  1536          // U_DIM + DU_DIM
#define YD   1024          // Y_DIM
#define DT   0.01f         // delta

// ---------------------------------------------------------------------------
// fp32 WMMA 16x16x4 (wave32).  VGPR layouts per CDNA5 ISA §7.12.2:
//  A (16x4):  lanes 0-15 -> M=lane, VGPR0/1 = K 0/1; lanes 16-31 -> K 2/3
//  B (4x16):  VGPR v, lanes 0-15 -> K=v, N=lane; lanes 16-31 -> K=v+2
//  C (16x16): VGPR v, lanes 0-15 -> M=v, N=lane; lanes 16-31 -> M=v+8
// ---------------------------------------------------------------------------
__device__ __forceinline__ v8f wmma4(v2f a, v2f b, v8f c) {
  return __builtin_amdgcn_wmma_f32_16x16x4_f32(false, a, false, b, (short)0, c,
                                               false, false);
}

// ---------------------------------------------------------------------------
// 64x64-per-wave NN GEMM body.  Two live base pointers; all row/tile
// displacements are immediate offsets.  16 WMMA : 12 loads per K-step.
// ---------------------------------------------------------------------------
#define GEMM64_BODY(Aptr, lda, Bptr, ldb, Kdim)                                 \
  v8f acc[4][4] = {};                                                           \
  {                                                                             \
    const float* ap = (Aptr) + (size_t)(i0 + (lane & 15)) * (lda) +             \
                      ((lane >> 4) << 1);                                       \
    const float* bp = (Bptr) + (size_t)((lane >> 4) << 1) * (ldb) + j0 +        \
                      (lane & 15);                                              \
    for (int k = 0; k < (Kdim); k += 4) {                                       \
      v2f af[4];                                                                \
      _Pragma("unroll")                                                         \
      for (int r = 0; r < 4; ++r) {                                             \
        af[r].x = ap[(size_t)(16 * r) * (lda)];                                 \
        af[r].y = ap[(size_t)(16 * r) * (lda) + 1];                             \
      }                                                                         \
      _Pragma("unroll")                                                         \
      for (int t = 0; t < 4; ++t) {                                             \
        v2f bf;                                                                 \
        bf.x = bp[16 * t];                                                      \
        bf.y = bp[16 * t + (ldb)];                                              \
        _Pragma("unroll")                                                       \
        for (int r = 0; r < 4; ++r) acc[r][t] = wmma4(af[r], bf, acc[r][t]);    \
      }                                                                         \
      ap += 4;                                                                  \
      bp += (size_t)4 * (ldb);                                                  \
    }                                                                           \
  }

#define STORE64_C(Cptr, ldc)                                                    \
  {                                                                             \
    float* cp = (Cptr) + (size_t)(i0 + ((lane >> 4) << 3)) * (ldc) + j0 +       \
                (lane & 15);                                                    \
    _Pragma("unroll")                                                           \
    for (int r = 0; r < 4; ++r)                                                 \
      _Pragma("unroll")                                                         \
      for (int t = 0; t < 4; ++t)                                               \
        _Pragma("unroll")                                                       \
        for (int v = 0; v < 8; ++v)                                             \
          cp[(size_t)(16 * r + v) * (ldc) + 16 * t] = acc[r][t][v];             \
  }

// ---------------------------------------------------------------------------
// X = delta * 0.5 * (W - W^T)
// ---------------------------------------------------------------------------
__global__ void k_build_x(const float* __restrict__ W, float* __restrict__ X) {
  int c = blockIdx.x * 16 + threadIdx.x;
  int r = blockIdx.y * 16 + threadIdx.y;
  X[(size_t)r * HD + c] = 0.5f * DT * (W[(size_t)r * HD + c] - W[(size_t)c * HD + r]);
}

// ---------------------------------------------------------------------------
// out[C x R] = in[R x C]^T   (32x32 LDS tile, coalesced both directions)
// ---------------------------------------------------------------------------
__global__ void k_transpose(const float* __restrict__ in, float* __restrict__ out,
                            int R, int C) {
  __shared__ float t[32][33];
  int c = blockIdx.x * 32 + threadIdx.x;
  int r = blockIdx.y * 32 + threadIdx.y;
#pragma unroll
  for (int i = 0; i < 32; i += 8)
    t[threadIdx.y + i][threadIdx.x] = in[(size_t)(r + i) * C + c];
  __syncthreads();
  int oc = blockIdx.y * 32 + threadIdx.x;   // output col = input row
  int orow = blockIdx.x * 32 + threadIdx.y; // output row = input col
#pragma unroll
  for (int i = 0; i < 32; i += 8)
    out[(size_t)(orow + i) * R + oc] = t[threadIdx.x][threadIdx.y + i];
}

// ---------------------------------------------------------------------------
// X2 = X * X   (2048^3 fp32 WMMA GEMM, wave tile 64x64)
// ---------------------------------------------------------------------------
__global__ void __launch_bounds__(128, 1)
k_gemm_x2(const float* __restrict__ X, float* __restrict__ X2) {
  int lane = threadIdx.x & 31, wave = threadIdx.x >> 5;
  int i0 = (blockIdx.y * 4 + wave) * 64;
  int j0 = blockIdx.x * 64;
  GEMM64_BODY(X, HD, X, HD, HD)
  STORE64_C(X2, HD)
}

// ---------------------------------------------------------------------------
// Fused:  E  = I + X + X2 * P,       P  = 1/2 I + X/6  + X2/24
//         MT = DT*(I - X/2 + X2*Q),  Q  = 1/6 I - X/24 + X2/120   (= M^T)
// Wave tile 32x64, dual accumulator banks; bp/bq synthesized from 4 loads and
// fed to 2 row-fragments each -> 4 WMMA per column tile per K-step.
// ---------------------------------------------------------------------------
__global__ void __launch_bounds__(128, 1)
k_gemm_em(const float* __restrict__ X, const float* __restrict__ X2,
          float* __restrict__ E, float* __restrict__ MT) {
  int lane = threadIdx.x & 31, wave = threadIdx.x >> 5;
  int i0 = (blockIdx.y * 4 + wave) * 32;
  int j0 = blockIdx.x * 64;
  int col = j0 + (lane & 15);
  v8f accP[2][4] = {}, accQ[2][4] = {};
  const float* ap = X2 + (size_t)(i0 + (lane & 15)) * HD + ((lane >> 4) << 1);
  const float* xp = X  + (size_t)((lane >> 4) << 1) * HD + col;
  const float* sp = X2 + (size_t)((lane >> 4) << 1) * HD + col;
  for (int k = 0; k < HD; k += 4) {
    int kk = k + ((lane >> 4) << 1);
    v2f af[2];
#pragma unroll
    for (int r = 0; r < 2; ++r) {
      af[r].x = ap[(size_t)(16 * r) * HD];
      af[r].y = ap[(size_t)(16 * r) * HD + 1];
    }
#pragma unroll
    for (int t = 0; t < 4; ++t) {
      float x0 = xp[16 * t], x1 = xp[16 * t + HD];
      float s0 = sp[16 * t], s1 = sp[16 * t + HD];
      int   ct = col + 16 * t;
      float d0 = (kk == ct) ? 1.0f : 0.0f;
      float d1 = (kk + 1 == ct) ? 1.0f : 0.0f;
      v2f bp_, bq_;
      bp_.x = 0.5f * d0 + x0 * (1.0f / 6.0f)  + s0 * (1.0f / 24.0f);
      bp_.y = 0.5f * d1 + x1 * (1.0f / 6.0f)  + s1 * (1.0f / 24.0f);
      bq_.x = d0 * (1.0f / 6.0f) - x0 * (1.0f / 24.0f) + s0 * (1.0f / 120.0f);
      bq_.y = d1 * (1.0f / 6.0f) - x1 * (1.0f / 24.0f) + s1 * (1.0f / 120.0f);
#pragma unroll
      for (int r = 0; r < 2; ++r) {
        accP[r][t] = wmma4(af[r], bp_, accP[r][t]);
        accQ[r][t] = wmma4(af[r], bq_, accQ[r][t]);
      }
    }
    ap += 4;
    xp += (size_t)4 * HD;
    sp += (size_t)4 * HD;
  }
  // Epilogue: add I + X terms, scale M^T by delta.
  {
    const float* xe = X + (size_t)(i0 + ((lane >> 4) << 3)) * HD + col;
    float* ep = E  + (size_t)(i0 + ((lane >> 4) << 3)) * HD + col;
    float* mp = MT + (size_t)(i0 + ((lane >> 4) << 3)) * HD + col;
    int rbase = i0 + ((lane >> 4) << 3);
#pragma unroll
    for (int r = 0; r < 2; ++r)
#pragma unroll
      for (int t = 0; t < 4; ++t)
#pragma unroll
        for (int v = 0; v < 8; ++v) {
          int row = rbase + 16 * r + v;
          size_t off = (size_t)(16 * r + v) * HD + 16 * t;
          float d  = (row == col + 16 * t) ? 1.0f : 0.0f;
          float xv = xe[off];
          ep[off] = d + xv + accP[r][t][v];
          mp[off] = DT * (d - 0.5f * xv + accQ[r][t][v]);
        }
  }
}

// ---------------------------------------------------------------------------
// rec[j] = sum_k h[k] * E[j][k]    (wave-per-row GEMV, coalesced)
// ---------------------------------------------------------------------------
__global__ void k_rec(const float* __restrict__ E, const float* __restrict__ h,
                      float* __restrict__ rec) {
  int lane = threadIdx.x & 31, wave = threadIdx.x >> 5;
  int j = blockIdx.x * 8 + wave;
  float s = 0.0f;
  for (int k = lane; k < HD; k += 32) s += E[(size_t)j * HD + k] * h[k];
#pragma unroll
  for (int off = 16; off > 0; off >>= 1) s += __shfl_down(s, off, 32);
  if (lane == 0) rec[j] = s;
}

// ---------------------------------------------------------------------------
// b = concat([du, u]) @ Bw^T   (K-loop split at uniform du/u boundary; the
// B pointer runs continuously across both loops)
// ---------------------------------------------------------------------------
__global__ void __launch_bounds__(128, 1)
k_gemm_b(const float* __restrict__ du, const float* __restrict__ u,
         const float* __restrict__ BwT, float* __restrict__ bmat) {
  int lane = threadIdx.x & 31, wave = threadIdx.x >> 5;
  int i0 = (blockIdx.y * 4 + wave) * 64;
  int j0 = blockIdx.x * 64;
  v8f acc[4][4] = {};
  const float* bp = BwT + (size_t)((lane >> 4) << 1) * HD + j0 + (lane & 15);
  {
    const float* ap = du + (size_t)(i0 + (lane & 15)) * DUD + ((lane >> 4) << 1);
    for (int k = 0; k < DUD; k += 4) {
      v2f af[4];
#pragma unroll
      for (int r = 0; r < 4; ++r) {
        af[r].x = ap[(size_t)(16 * r) * DUD];
        af[r].y = ap[(size_t)(16 * r) * DUD + 1];
      }
#pragma unroll
      for (int t = 0; t < 4; ++t) {
        v2f bf; bf.x = bp[16 * t]; bf.y = bp[16 * t + HD];
#pragma unroll
        for (int r = 0; r < 4; ++r) acc[r][t] = wmma4(af[r], bf, acc[r][t]);
      }
      ap += 4;
      bp += (size_t)4 * HD;
    }
  }
  {
    const float* ap = u + (size_t)(i0 + (lane & 15)) * UD + ((lane >> 4) << 1);
    for (int k = 0; k < UD; k += 4) {
      v2f af[4];
#pragma unroll
      for (int r = 0; r < 4; ++r) {
        af[r].x = ap[(size_t)(16 * r) * UD];
        af[r].y = ap[(size_t)(16 * r) * UD + 1];
      }
#pragma unroll
      for (int t = 0; t < 4; ++t) {
        v2f bf; bf.x = bp[16 * t]; bf.y = bp[16 * t + HD];
#pragma unroll
        for (int r = 0; r < 4; ++r) acc[r][t] = wmma4(af[r], bf, acc[r][t]);
      }
      ap += 4;
      bp += (size_t)4 * HD;
    }
  }
  STORE64_C(bmat, HD)
}

// ---------------------------------------------------------------------------
// hn = b @ MT + rec[j]   (NN GEMM, rec broadcast fused in epilogue)
// ---------------------------------------------------------------------------
__global__ void __launch_bounds__(128, 1)
k_gemm_hn(const float* __restrict__ bmat, const float* __restrict__ MT,
          const float* __restrict__ rec, float* __restrict__ hn) {
  int lane = threadIdx.x & 31, wave = threadIdx.x >> 5;
  int i0 = (blockIdx.y * 4 + wave) * 64;
  int j0 = blockIdx.x * 64;
  GEMM64_BODY(bmat, HD, MT, HD, HD)
  {
    float* cp = hn + (size_t)(i0 + ((lane >> 4) << 3)) * HD + j0 + (lane & 15);
    const float* rp = rec + j0 + (lane & 15);
#pragma unroll
    for (int t = 0; t < 4; ++t) {
      float rv = rp[16 * t];
#pragma unroll
      for (int r = 0; r < 4; ++r)
#pragma unroll
        for (int v = 0; v < 8; ++v)
          cp[(size_t)(16 * r + v) * HD + 16 * t] = acc[r][t][v] + rv;
    }
  }
}

// ---------------------------------------------------------------------------
// y = hn @ Cw^T   (NN GEMM with pre-transposed CwT) -> d_out
// ---------------------------------------------------------------------------
__global__ void __launch_bounds__(128, 1)
k_gemm_y(const float* __restrict__ hn, const float* __restrict__ CwT,
         float* __restrict__ y) {
  int lane = threadIdx.x & 31, wave = threadIdx.x >> 5;
  int i0 = (blockIdx.y * 4 + wave) * 64;
  int j0 = blockIdx.x * 64;
  GEMM64_BODY(hn, HD, CwT, YD, HD)
  STORE64_C(y, YD)
}

// ---------------------------------------------------------------------------
extern "C" void kernel_launch(void* const* d_in, const int* in_sizes, int n_in,
                              void* d_out, int out_size, void* d_ws, size_t ws_size,
                              hipStream_t stream) {
  const float* u  = (const float*)d_in[0];   // (512, 1024)
  const float* du = (const float*)d_in[1];   // (512, 512)
  const float* W  = (const float*)d_in[2];   // (2048, 2048)
  const float* Bw = (const float*)d_in[3];   // (2048, 1536)
  const float* Cw = (const float*)d_in[4];   // (1024, 2048)
  const float* h  = (const float*)d_in[5];   // (1, 2048)
  float* y = (float*)d_out;                  // (512, 1024)

  // Workspace layout (floats)
  float* X    = (float*)d_ws;                       // 2048*2048
  float* X2   = X    + (size_t)HD * HD;             // 2048*2048
  float* E    = X2   + (size_t)HD * HD;             // 2048*2048
  float* MT   = E    + (size_t)HD * HD;             // 2048*2048
  float* BwT  = MT   + (size_t)HD * HD;             // 1536*2048
  float* CwT  = BwT  + (size_t)KIN * HD;            // 2048*1024
  float* bmat = CwT  + (size_t)HD * YD;             // 512*2048
  float* hn   = bmat + (size_t)BSZ * HD;            // 512*2048
  float* rec  = hn   + (size_t)BSZ * HD;            // 2048

  // 1. X = 0.5*DT*(W - W^T)
  k_build_x<<<dim3(HD / 16, HD / 16), dim3(16, 16), 0, stream>>>(W, X);
  // 2. Pre-transpose weights so all GEMMs are NN / coalesced
  k_transpose<<<dim3(KIN / 32, HD / 32), dim3(32, 8), 0, stream>>>(Bw, BwT, HD, KIN);
  k_transpose<<<dim3(HD / 32, YD / 32), dim3(32, 8), 0, stream>>>(Cw, CwT, YD, HD);
  // 3. X2 = X*X
  k_gemm_x2<<<dim3(HD / 64, HD / 256), 128, 0, stream>>>(X, X2);
  // 4. E = expm(X) (4th order), MT = M^T = DT*phi1(X)^T
  k_gemm_em<<<dim3(HD / 64, HD / 128), 128, 0, stream>>>(X, X2, E, MT);
  // 5. rec = h @ E^T
  k_rec<<<HD / 8, 256, 0, stream>>>(E, h, rec);
  // 6. b = [du, u] @ Bw^T
  k_gemm_b<<<dim3(HD / 64, BSZ / 256), 128, 0, stream>>>(du, u, BwT, bmat);
  // 7. hn = b @ M^T + rec
  k_gemm_hn<<<dim3(HD / 64, BSZ / 256), 128, 0, stream>>>(bmat, MT, rec, hn);
  // 8. y = hn @ Cw^T
  k_gemm_y<<<dim3(YD / 64, BSZ / 256), 128, 0, stream>>>(hn, CwT, y);
}